// TsStddev_29257317220816
// MI455X (gfx1250) — compile-verified
//
#include <hip/hip_runtime.h>
#include <stdint.h>

namespace {
constexpr int kS      = 2048;  // samples per (b, f) row
constexpr int kWin    = 10;    // MIN
constexpr int kStride = 10;    // STRIDE
constexpr int kT      = 204;   // (2048 - 10) / 10 + 1 windows per row
constexpr int kBlock  = 256;   // 8 waves (wave32)
}

// One block per (b, f) row.
// Phase 1: async-copy the whole 8 KB row to LDS with two coalesced
//          global_load_async_to_lds_b128 per lane (CDNA5 async VMEM->LDS path,
//          ASYNCcnt-tracked, no VGPR staging). Input is streamed exactly once,
//          so mark the loads non-temporal (th:TH_LOAD_NT) to keep the 192 MB
//          L2 from caching a 377 MB one-shot stream.
// Phase 2: lanes 0..203 compute the sample stddev of their 10-wide window
//          straight out of LDS (stride-40B reads are bank-conflict-free on the
//          64-bank LDS) and store one f32 each, also non-temporal (write-once).
__global__ __launch_bounds__(kBlock) void ts_stddev_kernel(
    const float* __restrict__ x, float* __restrict__ out) {
  __shared__ __align__(16) float smem[kS];

  const int row = blockIdx.x;
  const int tid = threadIdx.x;

  // LDS byte offset of this lane's destination (low 32 bits of the flat
  // pointer to a __shared__ object are the LDS offset on AMDGPU).
  unsigned lds = (unsigned)(uintptr_t)(&smem[0]) + (unsigned)(tid * 16);
  // Global source address for this lane's first 16-byte chunk of the row.
  unsigned long long gaddr = (unsigned long long)(uintptr_t)x +
                             (unsigned long long)row * (unsigned long long)(kS * 4) +
                             (unsigned long long)(tid * 16);

  // 256 lanes x 16 B = 4096 B per instruction; offset:4096 applies to BOTH the
  // LDS destination and the global source (CDNA5 ISA), covering
  // the second half of the 8192 B row. Drain this wave's ASYNCcnt before the
  // workgroup barrier so every wave's LDS data is visible afterwards.
  asm volatile(
      "global_load_async_to_lds_b128 %0, %1, off th:TH_LOAD_NT\n\t"
      "global_load_async_to_lds_b128 %0, %1, off offset:4096 th:TH_LOAD_NT\n\t"
      "s_wait_asynccnt 0x0"
      :
      : "v"(lds), "v"(gaddr)
      : "memory");
  __syncthreads();

  if (tid < kT) {
    const float* wp = &smem[tid * kStride];
    float v[kWin];
#pragma unroll
    for (int j = 0; j < kWin; ++j) v[j] = wp[j];

    // Pairwise (log-depth) sum for the mean.
    const float s01 = v[0] + v[1];
    const float s23 = v[2] + v[3];
    const float s45 = v[4] + v[5];
    const float s67 = v[6] + v[7];
    const float s89 = v[8] + v[9];
    const float s0123 = s01 + s23;
    const float s4567 = s45 + s67;
    const float mean = ((s0123 + s4567) + s89) * (1.0f / (float)kWin);

    // Centered sum of squares, two FMA chains to expose ILP.
    float ssa = 0.0f, ssb = 0.0f;
#pragma unroll
    for (int j = 0; j < kWin; j += 2) {
      const float da = v[j] - mean;
      const float db = v[j + 1] - mean;
      ssa = __builtin_fmaf(da, da, ssa);
      ssb = __builtin_fmaf(db, db, ssb);
    }
    const float r = sqrtf((ssa + ssb) * (1.0f / (float)(kWin - 1)));
    __builtin_nontemporal_store(r, out + (size_t)row * kT + tid);
  }
}

extern "C" void kernel_launch(void* const* d_in, const int* in_sizes, int n_in,
                              void* d_out, int out_size, void* d_ws, size_t ws_size,
                              hipStream_t stream) {
  const float* x = (const float*)d_in[0];
  float* out     = (float*)d_out;
  const int rows = in_sizes[0] / kS;  // 128 * 360 = 46080 rows
  ts_stddev_kernel<<<rows, kBlock, 0, stream>>>(x, out);
}